// RecurrentGATTrackerV2_70385924047134
// MI455X (gfx1250) — compile-verified
//
#include <hip/hip_runtime.h>
#include <hip/hip_bf16.h>
#include <stdint.h>

// ---------------------------------------------------------------------------
// RecurrentGATTrackerV2 for gfx1250 (MI455X).
// Dense GEMMs: V_WMMA_F32_16X16X4_F32, one 16x64 strip per wave (A fragment
// reused across 4 column tiles -> 4 WMMAs per K-step).
// Edge phases: 16 lanes per edge, 4 channels/lane -> coalesced b128 gathers
// and coalesced atomics into the 192MB-L2-resident node tables.
// ---------------------------------------------------------------------------

typedef __attribute__((ext_vector_type(2))) float v2f;
typedef __attribute__((ext_vector_type(8))) float v8f;

#define HID   64
#define HEADS 4
#define CH    16
#define EDIM  6
#define NEG   0.2f

// ---- monotonic float <-> uint key for atomicMax-based segment max ---------
__device__ __forceinline__ unsigned fkey(float f) {
    unsigned u = __float_as_uint(f);
    return (u & 0x80000000u) ? ~u : (u | 0x80000000u);
}
__device__ __forceinline__ float funkey(unsigned k) {
    unsigned u = (k & 0x80000000u) ? (k & 0x7FFFFFFFu) : ~k;
    return __uint_as_float(u);
}

// ---------------------------------------------------------------------------
// fp32 WMMA GEMM:  C[M x Nc] = act(A[M x K] @ W[K x Nc] + bias)
// One wave owns a 16x64 strip: 4 accumulators, A fragment loaded once per
// K-step and fed to 4 WMMAs. M %16==0, K %4==0, Nc %64==0.
// ---------------------------------------------------------------------------
__global__ void gemm_wmma_f32(const float* __restrict__ A,
                              const float* __restrict__ W,
                              const float* __restrict__ bias,
                              float* __restrict__ C,
                              int M, int K, int Nc, int act) {
    const int lane = threadIdx.x & 31;
    const int wave = threadIdx.x >> 5;
    const int nStripsN = Nc >> 6;                  // strips of 64 columns
    const int nStrips  = (M >> 4) * nStripsN;
    const int strip = blockIdx.x * (blockDim.x >> 5) + wave;
    if (strip >= nStrips) return;                  // wave-uniform -> EXEC full

    const int tm = strip / nStripsN, ts = strip % nStripsN;
    const int r0 = tm << 4, c0 = ts << 6;
    const int hi = lane >> 4;                      // 0:(k0,k0+1) 1:(k0+2,k0+3)
    const int lo = lane & 15;

    const float* Arow = A + (size_t)(r0 + lo) * K;
    const float* Wcol = W + (c0 + lo);

    v8f acc0 = {}, acc1 = {}, acc2 = {}, acc3 = {};
    for (int k0 = 0; k0 < K; k0 += 4) {
        const int ka = k0 + (hi << 1);
        v2f a;
        a.x = Arow[ka];
        a.y = Arow[ka + 1];
        const float* w0 = Wcol + (size_t)ka * Nc;
        const float* w1 = Wcol + (size_t)(ka + 1) * Nc;
        v2f b0, b1, b2, b3;
        b0.x = w0[0];   b0.y = w1[0];
        b1.x = w0[16];  b1.y = w1[16];
        b2.x = w0[32];  b2.y = w1[32];
        b3.x = w0[48];  b3.y = w1[48];
        acc0 = __builtin_amdgcn_wmma_f32_16x16x4_f32(false, a, false, b0, (short)0, acc0, false, false);
        acc1 = __builtin_amdgcn_wmma_f32_16x16x4_f32(false, a, false, b1, (short)0, acc1, false, false);
        acc2 = __builtin_amdgcn_wmma_f32_16x16x4_f32(false, a, false, b2, (short)0, acc2, false, false);
        acc3 = __builtin_amdgcn_wmma_f32_16x16x4_f32(false, a, false, b3, (short)0, acc3, false, false);
    }

    float* Crow = C + (size_t)(r0 + hi * 8) * Nc + (c0 + lo);
#pragma unroll
    for (int t = 0; t < 4; ++t) {
        const v8f acc = (t == 0) ? acc0 : (t == 1) ? acc1 : (t == 2) ? acc2 : acc3;
        const float bv = bias ? bias[c0 + t * 16 + lo] : 0.0f;
#pragma unroll
        for (int v = 0; v < 8; ++v) {
            float xv = acc[v] + bv;
            if (act == 1) xv = fmaxf(xv, 0.0f);
            Crow[(size_t)v * Nc + t * 16] = xv;
        }
    }
}

// ---------------------------------------------------------------------------
// Feature assembly: feat[N x 32] = [x(7) | pos(3) | type_emb[nt](8) | 0...]
// ---------------------------------------------------------------------------
__global__ void build_feat(int N, const float* __restrict__ x,
                           const float* __restrict__ pos,
                           const int* __restrict__ nt,
                           const float* __restrict__ temb,
                           float* __restrict__ feat) {
    int i = blockIdx.x * blockDim.x + threadIdx.x;
    if (i >= N) return;
    float* f = feat + (size_t)i * 32;
#pragma unroll
    for (int k = 0; k < 7; ++k) f[k] = x[(size_t)i * 7 + k];
#pragma unroll
    for (int k = 0; k < 3; ++k) f[7 + k] = pos[(size_t)i * 3 + k];
    const int t = nt[i];
#pragma unroll
    for (int k = 0; k < 8; ++k) f[10 + k] = temb[t * 8 + k];
#pragma unroll
    for (int k = 18; k < 32; ++k) f[k] = 0.0f;
}

// Zero-pad enc_W1 (18x64) into 32x64
__global__ void pad_w1(const float* __restrict__ W1, float* __restrict__ Wp) {
    int t = blockIdx.x * blockDim.x + threadIdx.x;
    if (t >= 32 * 64) return;
    Wp[t] = ((t >> 6) < 18) ? W1[t] : 0.0f;
}

// ---------------------------------------------------------------------------
// deg + loop_attr accumulation, then normalization
// ---------------------------------------------------------------------------
__global__ void deg_loop(int E, const int* __restrict__ ei,
                         const float* __restrict__ eattr,
                         float* __restrict__ deg, float* __restrict__ loopA) {
    int e = blockIdx.x * blockDim.x + threadIdx.x;
    if (e >= E) return;
    const int d = ei[E + e];
    atomicAdd(deg + d, 1.0f);
#pragma unroll
    for (int k = 0; k < EDIM; ++k)
        atomicAdd(loopA + (size_t)d * EDIM + k, eattr[(size_t)e * EDIM + k]);
}

__global__ void loop_div(int N, const float* __restrict__ deg,
                         float* __restrict__ loopA) {
    int i = blockIdx.x * blockDim.x + threadIdx.x;
    if (i >= N) return;
    const float s = 1.0f / fmaxf(deg[i], 1.0f);
#pragma unroll
    for (int k = 0; k < EDIM; ++k) loopA[(size_t)i * EDIM + k] *= s;
}

// ---------------------------------------------------------------------------
// GATv2 edge pass 1: e = att . leaky_relu(xl[src] + xr[dst] + eattr@We)
// 16 lanes per edge, 4 channels per lane; coalesced b128 gathers; per-head
// reduction via shfl_xor inside aligned quads; atomicMax per (dst, head).
// ---------------------------------------------------------------------------
__global__ void edge_score(int E, int N, const int* __restrict__ ei,
                           const float* __restrict__ eattr,
                           const float* __restrict__ loopA,
                           const float* __restrict__ xl,
                           const float* __restrict__ xr,
                           const float* __restrict__ We,   // 6 x 64
                           const float* __restrict__ att,  // 4 x 16 flat
                           float* __restrict__ eOut,       // (E+N) x 4
                           unsigned* __restrict__ maxb) {
    __shared__ float sWe[EDIM * HID];
    __shared__ float sAtt[HID];
    for (int t = threadIdx.x; t < EDIM * HID; t += blockDim.x) sWe[t] = We[t];
    if (threadIdx.x < HID) sAtt[threadIdx.x] = att[threadIdx.x];
    __syncthreads();

    const int Ea = E + N;
    const int gid = blockIdx.x * blockDim.x + threadIdx.x;
    const int e = gid >> 4;          // edge per 16-lane group
    const int c = gid & 15;          // channel group: handles j = 4c..4c+3
    if (e >= Ea) return;

    int s, d;
    const float* ea;
    if (e < E) { s = ei[e]; d = ei[E + e]; ea = eattr + (size_t)e * EDIM; }
    else       { s = d = e - E;            ea = loopA + (size_t)(e - E) * EDIM; }

    float at[EDIM];
#pragma unroll
    for (int k = 0; k < EDIM; ++k) at[k] = ea[k];

    const float4 xv = ((const float4*)xl)[(size_t)s * 16 + c];
    const float4 rv = ((const float4*)xr)[(size_t)d * 16 + c];
    const float xs[4] = {xv.x, xv.y, xv.z, xv.w};
    const float rs[4] = {rv.x, rv.y, rv.z, rv.w};

    float part = 0.0f;
#pragma unroll
    for (int q = 0; q < 4; ++q) {
        const int j = 4 * c + q;
        float v = xs[q] + rs[q];
#pragma unroll
        for (int k = 0; k < EDIM; ++k) v += at[k] * sWe[k * HID + j];
        v = (v > 0.0f) ? v : NEG * v;
        part += v * sAtt[j];
    }
    // reduce 4 lanes of the quad (channels of one head)
    part += __shfl_xor(part, 1);
    part += __shfl_xor(part, 2);
    if ((c & 3) == 0) {
        const int h = c >> 2;
        eOut[(size_t)e * HEADS + h] = part;
        atomicMax(maxb + (size_t)d * HEADS + h, fkey(part));
    }
}

// pass 2: one thread per (edge, head): ex = exp(e - max[dst]); sum atomics
__global__ void edge_exp(int E, int N, const int* __restrict__ ei,
                         float* __restrict__ eio,
                         const unsigned* __restrict__ maxb,
                         float* __restrict__ sumb) {
    const int Ea = E + N;
    const int idx = blockIdx.x * blockDim.x + threadIdx.x;
    const int e = idx >> 2, h = idx & 3;
    if (e >= Ea) return;
    const int d = (e < E) ? ei[E + e] : (e - E);
    const unsigned k = maxb[(size_t)d * HEADS + h];
    float m = (k == 0u) ? 0.0f : funkey(k);
    if (!isfinite(m)) m = 0.0f;
    const float ex = __expf(eio[(size_t)e * HEADS + h] - m);
    eio[(size_t)e * HEADS + h] = ex;
    atomicAdd(sumb + (size_t)d * HEADS + h, ex);
}

// pass 3: 16 lanes per edge: alpha = ex/(sum+eps) (stored by quad leaders),
// msg = xl[src] * alpha scattered with coalesced atomics (4 consecutive/lane)
__global__ void edge_msg(int E, int N, const int* __restrict__ ei,
                         float* __restrict__ alphio,
                         const float* __restrict__ sumb,
                         const float* __restrict__ xl,
                         float* __restrict__ dest) {
    const int Ea = E + N;
    const int gid = blockIdx.x * blockDim.x + threadIdx.x;
    const int e = gid >> 4;
    const int c = gid & 15;
    if (e >= Ea) return;
    int s, d;
    if (e < E) { s = ei[e]; d = ei[E + e]; } else { s = d = e - E; }

    const int h = c >> 2;
    const float al = alphio[(size_t)e * HEADS + h] /
                     (sumb[(size_t)d * HEADS + h] + 1e-16f);
    if ((c & 3) == 0) alphio[(size_t)e * HEADS + h] = al;  // same-wave lockstep

    const float4 xv = ((const float4*)xl)[(size_t)s * 16 + c];
    float* dp = dest + (size_t)d * HID + 4 * c;
    atomicAdd(dp + 0, xv.x * al);
    atomicAdd(dp + 1, xv.y * al);
    atomicAdd(dp + 2, xv.z * al);
    atomicAdd(dp + 3, xv.w * al);
}

// h1 = relu(out + bias) in place
__global__ void epilogue_relu(size_t n, const float* __restrict__ bias,
                              float* __restrict__ io) {
    size_t i = (size_t)blockIdx.x * blockDim.x + threadIdx.x;
    if (i >= n) return;
    io[i] = fmaxf(io[i] + bias[i & (HID - 1)], 0.0f);
}

// h2 = acc + bias in place; rows >= T also copied to new_full region
__global__ void epilogue_h2(size_t n, int T, const float* __restrict__ bias,
                            float* __restrict__ io, float* __restrict__ newfull) {
    size_t i = (size_t)blockIdx.x * blockDim.x + threadIdx.x;
    if (i >= n) return;
    const float v = io[i] + bias[i & (HID - 1)];
    io[i] = v;
    if ((int)(i >> 6) >= T) newfull[i] = v;
}

// GRU gates -> new_tracks written into new_full[0:T]
__global__ void gru_gate(int T, const float* __restrict__ gi,
                         const float* __restrict__ gh,
                         const float* __restrict__ hidden,
                         float* __restrict__ newfull) {
    int idx = blockIdx.x * blockDim.x + threadIdx.x;
    if (idx >= T * HID) return;
    const int t = idx >> 6, j = idx & (HID - 1);
    const float* git = gi + (size_t)t * 3 * HID;
    const float* ght = gh + (size_t)t * 3 * HID;
    const float r = 1.0f / (1.0f + __expf(-(git[j] + ght[j])));
    const float z = 1.0f / (1.0f + __expf(-(git[HID + j] + ght[HID + j])));
    const float nn = tanhf(git[2 * HID + j] + r * ght[2 * HID + j]);
    newfull[idx] = (1.0f - z) * nn + z * hidden[idx];
}

// decoder stage 2 (64 -> 7) + residual position fixups
__global__ void dec2_final(int N, const float* __restrict__ d1,
                           const float* __restrict__ W2,  // 64 x 7
                           const float* __restrict__ b2,  // 7
                           const float* __restrict__ pos,
                           const float* __restrict__ x,
                           float* __restrict__ outF) {
    __shared__ float sW[HID * 7 + 7];
    for (int t = threadIdx.x; t < HID * 7 + 7; t += blockDim.x)
        sW[t] = (t < HID * 7) ? W2[t] : b2[t - HID * 7];
    __syncthreads();
    int i = blockIdx.x * blockDim.x + threadIdx.x;
    if (i >= N) return;
    const float* row = d1 + (size_t)i * HID;
    float o[7];
#pragma unroll
    for (int c = 0; c < 7; ++c) o[c] = sW[HID * 7 + c];
    for (int j = 0; j < HID; ++j) {
        const float v = row[j];
#pragma unroll
        for (int c = 0; c < 7; ++c) o[c] += v * sW[j * 7 + c];
    }
#pragma unroll
    for (int c = 0; c < 7; ++c) {
        float v = o[c];
        if (c < 3)      v += pos[(size_t)i * 3 + c];
        else if (c < 6) v += x[(size_t)i * 7 + (c - 3)];
        outF[(size_t)i * 7 + c] = v;
    }
}

// ---------------------------------------------------------------------------
static inline int cdiv(long long a, int b) { return (int)((a + b - 1) / b); }

extern "C" void kernel_launch(void* const* d_in, const int* in_sizes, int n_in,
                              void* d_out, int out_size, void* d_ws, size_t ws_size,
                              hipStream_t stream) {
    const float* x        = (const float*)d_in[0];
    const float* pos      = (const float*)d_in[1];
    const int*   ntype    = (const int*)  d_in[2];
    const int*   ei       = (const int*)  d_in[3];
    const float* eattr    = (const float*)d_in[4];
    const float* hidden   = (const float*)d_in[6];
    const float* temb     = (const float*)d_in[7];
    const float* enc_W1   = (const float*)d_in[8];
    const float* enc_b1   = (const float*)d_in[9];
    const float* enc_W2   = (const float*)d_in[10];
    const float* enc_b2   = (const float*)d_in[11];
    const float* gru_Wi   = (const float*)d_in[12];
    const float* gru_bi   = (const float*)d_in[13];
    const float* gru_Wh   = (const float*)d_in[14];
    const float* gru_bh   = (const float*)d_in[15];
    const float* dec_W1   = (const float*)d_in[16];
    const float* dec_b1   = (const float*)d_in[17];
    const float* dec_W2   = (const float*)d_in[18];
    const float* dec_b2   = (const float*)d_in[19];
    const float* g1_Wl    = (const float*)d_in[20];
    const float* g1_bl    = (const float*)d_in[21];
    const float* g1_Wr    = (const float*)d_in[22];
    const float* g1_br    = (const float*)d_in[23];
    const float* g1_We    = (const float*)d_in[24];
    const float* g1_att   = (const float*)d_in[25];
    const float* g1_bias  = (const float*)d_in[26];
    const float* g2_Wl    = (const float*)d_in[27];
    const float* g2_bl    = (const float*)d_in[28];
    const float* g2_Wr    = (const float*)d_in[29];
    const float* g2_br    = (const float*)d_in[30];
    const float* g2_We    = (const float*)d_in[31];
    const float* g2_att   = (const float*)d_in[32];
    const float* g2_bias  = (const float*)d_in[33];

    const int N  = in_sizes[0] / 7;
    const int E  = in_sizes[4] / EDIM;
    const int T  = in_sizes[6] / HID;
    const int Ea = E + N;

    // workspace layout (floats)
    float* ws    = (float*)d_ws;
    float* feat  = ws;                         // N*32
    float* Wp    = feat + (size_t)N * 32;      // 32*64
    float* Abuf  = Wp + 32 * 64;               // N*64 (t1, then dec hidden)
    float* H     = Abuf + (size_t)N * HID;     // N*64 (h -> layer2 acc -> h2)
    float* XL    = H + (size_t)N * HID;        // N*64 (xl, then GRU gi)
    float* XR    = XL + (size_t)N * HID;       // N*64 (xr, then GRU gh)
    float* OUT   = XR + (size_t)N * HID;       // N*64 (layer1 acc -> h1)
    float* deg   = OUT + (size_t)N * HID;      // N
    float* loopA = deg + N;                    // N*6
    unsigned* maxb = (unsigned*)(loopA + (size_t)N * EDIM); // N*4
    float*    sumb = (float*)(maxb + (size_t)N * HEADS);    // N*4

    // output layout
    float* outF    = (float*)d_out;             // N*7
    float* newfull = outF + (size_t)N * 7;      // N*64
    float* a1      = newfull + (size_t)N * HID; // Ea*4
    float* a2      = a1 + (size_t)Ea * HEADS;   // Ea*4

    const int BT = 256;

    // 1) features + padded W1
    build_feat<<<cdiv(N, BT), BT, 0, stream>>>(N, x, pos, ntype, temb, feat);
    pad_w1<<<cdiv(32 * 64, BT), BT, 0, stream>>>(enc_W1, Wp);

    // 2) degree + loop_attr
    hipMemsetAsync(deg, 0, (size_t)N * 7 * sizeof(float), stream); // deg + loopA
    deg_loop<<<cdiv(E, BT), BT, 0, stream>>>(E, ei, eattr, deg, loopA);
    loop_div<<<cdiv(N, BT), BT, 0, stream>>>(N, deg, loopA);

    // 3) encoder: relu(feat @ Wp + b1) @ W2 + b2 -> H
    gemm_wmma_f32<<<cdiv((N / 16) * (HID / 64), 8), BT, 0, stream>>>(
        feat, Wp, enc_b1, Abuf, N, 32, HID, 1);
    gemm_wmma_f32<<<cdiv((N / 16) * (HID / 64), 8), BT, 0, stream>>>(
        Abuf, enc_W2, enc_b2, H, N, HID, HID, 0);

    // ---- GAT layer 1 ----
    gemm_wmma_f32<<<cdiv((N / 16) * (HID / 64), 8), BT, 0, stream>>>(
        H, g1_Wl, g1_bl, XL, N, HID, HID, 0);
    gemm_wmma_f32<<<cdiv((N / 16) * (HID / 64), 8), BT, 0, stream>>>(
        H, g1_Wr, g1_br, XR, N, HID, HID, 0);
    hipMemsetAsync(maxb, 0, (size_t)N * 2 * HEADS * sizeof(float), stream); // maxb+sumb
    hipMemsetAsync(OUT, 0, (size_t)N * HID * sizeof(float), stream);
    edge_score<<<cdiv((long long)Ea * 16, BT), BT, 0, stream>>>(
        E, N, ei, eattr, loopA, XL, XR, g1_We, g1_att, a1, maxb);
    edge_exp<<<cdiv((long long)Ea * 4, BT), BT, 0, stream>>>(
        E, N, ei, a1, maxb, sumb);
    edge_msg<<<cdiv((long long)Ea * 16, BT), BT, 0, stream>>>(
        E, N, ei, a1, sumb, XL, OUT);
    epilogue_relu<<<cdiv((long long)N * HID, BT), BT, 0, stream>>>(
        (size_t)N * HID, g1_bias, OUT);   // OUT = h1

    // ---- GAT layer 2 ----
    gemm_wmma_f32<<<cdiv((N / 16) * (HID / 64), 8), BT, 0, stream>>>(
        OUT, g2_Wl, g2_bl, XL, N, HID, HID, 0);
    gemm_wmma_f32<<<cdiv((N / 16) * (HID / 64), 8), BT, 0, stream>>>(
        OUT, g2_Wr, g2_br, XR, N, HID, HID, 0);
    hipMemsetAsync(maxb, 0, (size_t)N * 2 * HEADS * sizeof(float), stream);
    hipMemsetAsync(H, 0, (size_t)N * HID * sizeof(float), stream); // H = layer2 acc
    edge_score<<<cdiv((long long)Ea * 16, BT), BT, 0, stream>>>(
        E, N, ei, eattr, loopA, XL, XR, g2_We, g2_att, a2, maxb);
    edge_exp<<<cdiv((long long)Ea * 4, BT), BT, 0, stream>>>(
        E, N, ei, a2, maxb, sumb);
    edge_msg<<<cdiv((long long)Ea * 16, BT), BT, 0, stream>>>(
        E, N, ei, a2, sumb, XL, H);
    epilogue_h2<<<cdiv((long long)N * HID, BT), BT, 0, stream>>>(
        (size_t)N * HID, T, g2_bias, H, newfull); // H = h2; rows>=T -> newfull

    // ---- GRU on first T rows (XL, XR recycled as gi, gh) ----
    gemm_wmma_f32<<<cdiv((T / 16) * (3 * HID / 64), 8), BT, 0, stream>>>(
        H, gru_Wi, gru_bi, XL, T, HID, 3 * HID, 0);
    gemm_wmma_f32<<<cdiv((T / 16) * (3 * HID / 64), 8), BT, 0, stream>>>(
        hidden, gru_Wh, gru_bh, XR, T, HID, 3 * HID, 0);
    gru_gate<<<cdiv((long long)T * HID, BT), BT, 0, stream>>>(
        T, XL, XR, hidden, newfull);

    // ---- decoder ----
    gemm_wmma_f32<<<cdiv((N / 16) * (HID / 64), 8), BT, 0, stream>>>(
        newfull, dec_W1, dec_b1, Abuf, N, HID, HID, 1);
    dec2_final<<<cdiv(N, BT), BT, 0, stream>>>(N, Abuf, dec_W2, dec_b2,
                                               pos, x, outF);
}